// MortonDispatcher_23587960389769
// MI455X (gfx1250) — compile-verified
//
#include <hip/hip_runtime.h>
#include <hip/hip_bf16.h>
#include <math.h>

#define IN_DIM   1024
#define PROJ     8
#define BITS     6
#define RS_BLOCKS 256

typedef float v2f __attribute__((ext_vector_type(2)));
typedef float v8f __attribute__((ext_vector_type(8)));

// 64-bit xor-shuffle built from two 32-bit shuffles (wave32)
__device__ __forceinline__ unsigned long long shfl_xor_u64(unsigned long long v, int m) {
    int lo = __shfl_xor((int)(unsigned)(v & 0xffffffffull), m, 32);
    int hi = __shfl_xor((int)(unsigned)(v >> 32), m, 32);
    return ((unsigned long long)(unsigned)hi << 32) | (unsigned)lo;
}

// gfx1250 hardware tanh (V_TANH_F32, TRANS pipe) with libm fallback
__device__ __forceinline__ float hw_tanh(float x) {
#if __has_builtin(__builtin_amdgcn_tanhf)
    return __builtin_amdgcn_tanhf(x);
#else
    return tanhf(x);
#endif
}

// ---------------------------------------------------------------------------
// Fused gather + f32 WMMA projection + tanh/quantize + Morton code
// One wave per 16-row tile; 8 waves (256 threads) per block -> 128 rows/block.
// ---------------------------------------------------------------------------
__global__ __launch_bounds__(256) void morton_wmma_kernel(
    const float* __restrict__ flat, const long long* __restrict__ pos,
    const float* __restrict__ proj, unsigned long long* __restrict__ keys,
    unsigned* __restrict__ vals, int nSel)
{
    // proj staged padded to 16 columns (cols 8..15 = 0) => branch-free B loads
    __shared__ float sProj[IN_DIM * 16];              // 64 KB
    for (int i = threadIdx.x; i < IN_DIM * 16; i += 256) {
        int r = i >> 4, col = i & 15;
        sProj[i] = (col < PROJ) ? proj[r * PROJ + col] : 0.0f;
    }
    __syncthreads();

    const int lane  = threadIdx.x & 31;
    const int wave  = threadIdx.x >> 5;
    const int tileBase = (blockIdx.x * 8 + wave) * 16;
    if (tileBase + 15 >= nSel) return;

    const int n     = lane & 15;          // column (N) index for B/C fragments
    const int hi    = lane >> 4;          // lane half
    const int kkOff = hi * 2;             // K sub-offset for A/B fragments

    // A-fragment row: both halves cover M = 0..15, differing in K
    const long long prow = pos[tileBase + n];
    const float* aRow = flat + (size_t)prow * IN_DIM + kkOff;
    const float* bCol = sProj + n;        // lane-constant base into padded proj

    v8f c = {0.f, 0.f, 0.f, 0.f, 0.f, 0.f, 0.f, 0.f};

    #pragma unroll 8
    for (int k = 0; k < IN_DIM; k += 4) {
        // A 16x4 f32: lane half 0 holds K=k,k+1 ; half 1 holds K=k+2,k+3
        v2f a = *(const v2f*)(aRow + k);
        // B 4x16 f32: same K split per half; padded cols give 0 unconditionally
        v2f b;
        b.x = bCol[(k + kkOff)     * 16];
        b.y = bCol[(k + kkOff + 1) * 16];
        c = __builtin_amdgcn_wmma_f32_16x16x4_f32(
                /*neg_a=*/false, a, /*neg_b=*/false, b,
                /*c_mod=*/(short)0, c, /*reuse_a=*/false, /*reuse_b=*/false);
    }

    // C layout: lane holds column n of rows v + 8*hi (v = VGPR index 0..7)
    #pragma unroll
    for (int v = 0; v < 8; ++v) {
        float xs = 0.5f * (hw_tanh(c[v]) + 1.0f);
        xs = fminf(fmaxf(xs, 0.0f), 1.0f - 1e-6f);
        int q = (int)(xs * (float)((1 << BITS) - 1));
        unsigned long long s = 0ull;
        if (n < PROJ) {                   // cols 8..15 must not contribute bits
            #pragma unroll
            for (int b = 0; b < BITS; ++b)
                s |= (unsigned long long)((q >> b) & 1) << (n + PROJ * b);
        }
        // OR-reduce across the 16 lanes of this half (bits are disjoint per n)
        s |= shfl_xor_u64(s, 1);
        s |= shfl_xor_u64(s, 2);
        s |= shfl_xor_u64(s, 4);
        s |= shfl_xor_u64(s, 8);
        if (n == v) {                     // lanes 0..7 write rows 0..7, 16..23 rows 8..15
            int r = tileBase + v + hi * 8;
            keys[r] = s;
            vals[r] = (unsigned)pos[r];
        }
    }
}

// ---------------------------------------------------------------------------
// Stable LSD radix sort: 6 passes x 8-bit digits over 48-bit keys
// ---------------------------------------------------------------------------
__global__ __launch_bounds__(256) void rs_hist(const unsigned long long* __restrict__ keys,
                                               unsigned* __restrict__ hist, int n, int shift)
{
    __shared__ unsigned cnt[256];
    cnt[threadIdx.x] = 0;
    __syncthreads();
    const int per  = n / RS_BLOCKS;
    const int base = blockIdx.x * per;
    for (int i = threadIdx.x; i < per; i += 256) {
        unsigned d = (unsigned)(keys[base + i] >> shift) & 255u;
        atomicAdd(&cnt[d], 1u);
    }
    __syncthreads();
    // digit-major layout => linear scan is stable across blocks
    hist[threadIdx.x * RS_BLOCKS + blockIdx.x] = cnt[threadIdx.x];
}

__global__ __launch_bounds__(256) void rs_scan(unsigned* __restrict__ hist)
{
    __shared__ unsigned tmp[256];
    unsigned carry = 0;
    const int t = threadIdx.x;
    for (int chunk = 0; chunk < 256 * RS_BLOCKS; chunk += 256) {
        unsigned x = hist[chunk + t];
        tmp[t] = x;
        __syncthreads();
        #pragma unroll
        for (int off = 1; off < 256; off <<= 1) {
            unsigned y = (t >= off) ? tmp[t - off] : 0u;
            __syncthreads();
            tmp[t] += y;
            __syncthreads();
        }
        unsigned incl  = tmp[t];
        unsigned total = tmp[255];
        hist[chunk + t] = carry + incl - x;   // exclusive prefix
        carry += total;
        __syncthreads();
    }
}

__global__ __launch_bounds__(256) void rs_scatter(
    const unsigned long long* __restrict__ keysIn, const unsigned* __restrict__ valsIn,
    unsigned long long* __restrict__ keysOut, unsigned* __restrict__ valsOut,
    const unsigned* __restrict__ hist, int n, int shift)
{
    __shared__ unsigned waveCnt[8][256];
    __shared__ unsigned running[256];
    running[threadIdx.x] = 0;

    const int per  = n / RS_BLOCKS;
    const int base = blockIdx.x * per;
    const int lane = threadIdx.x & 31;
    const int wave = threadIdx.x >> 5;
    const unsigned ltMask = (lane == 0) ? 0u : (0xffffffffu >> (32 - lane));

    for (int c0 = 0; c0 < per; c0 += 256) {
        #pragma unroll
        for (int w = 0; w < 8; ++w) waveCnt[w][threadIdx.x] = 0;
        __syncthreads();

        const int i = base + c0 + threadIdx.x;
        unsigned long long key = keysIn[i];
        unsigned val = valsIn[i];
        unsigned d = (unsigned)(key >> shift) & 255u;

        // peer mask: lanes in this wave sharing my digit
        unsigned peers = 0xffffffffu;
        #pragma unroll
        for (int b = 0; b < 8; ++b) {
            unsigned bb = (unsigned)__ballot((d >> b) & 1u);
            peers &= ((d >> b) & 1u) ? bb : ~bb;
        }
        unsigned rank = __popc(peers & ltMask);         // stable rank within wave
        if (rank == 0) waveCnt[wave][d] = __popc(peers);
        __syncthreads();

        unsigned off = running[d];
        for (int w = 0; w < wave; ++w) off += waveCnt[w][d];
        unsigned gbase = hist[d * RS_BLOCKS + blockIdx.x];
        unsigned idx = gbase + off + rank;
        keysOut[idx] = key;
        valsOut[idx] = val;
        __syncthreads();

        unsigned tot = 0;
        #pragma unroll
        for (int w = 0; w < 8; ++w) tot += waveCnt[w][threadIdx.x];
        running[threadIdx.x] += tot;
        __syncthreads();
    }
}

__global__ __launch_bounds__(256) void cast_out_kernel(const unsigned* __restrict__ vals,
                                                       long long* __restrict__ out, int n)
{
    int i = blockIdx.x * 256 + threadIdx.x;
    if (i < n) out[i] = (long long)vals[i];
}

// ---------------------------------------------------------------------------
extern "C" void kernel_launch(void* const* d_in, const int* in_sizes, int n_in,
                              void* d_out, int out_size, void* d_ws, size_t ws_size,
                              hipStream_t stream)
{
    const float*      flat = (const float*)d_in[0];
    const long long*  pos  = (const long long*)d_in[1];
    const float*      proj = (const float*)d_in[2];
    const int nSel = in_sizes[1];

    unsigned char* ws = (unsigned char*)d_ws;
    unsigned long long* keysA = (unsigned long long*)ws;
    unsigned long long* keysB = keysA + nSel;
    unsigned* valsA = (unsigned*)(keysB + nSel);
    unsigned* valsB = valsA + nSel;
    unsigned* hist  = valsB + nSel;                       // 256 * RS_BLOCKS entries

    // 1) gather + project (f32 WMMA) + Morton codes
    morton_wmma_kernel<<<nSel / 128, 256, 0, stream>>>(flat, pos, proj, keysA, valsA, nSel);

    // 2) stable LSD radix sort, 6 passes of 8 bits (48-bit codes)
    unsigned long long* ki = keysA; unsigned long long* ko = keysB;
    unsigned* vi = valsA; unsigned* vo = valsB;
    for (int pass = 0; pass < 6; ++pass) {
        const int shift = pass * 8;
        rs_hist   <<<RS_BLOCKS, 256, 0, stream>>>(ki, hist, nSel, shift);
        rs_scan   <<<1,         256, 0, stream>>>(hist);
        rs_scatter<<<RS_BLOCKS, 256, 0, stream>>>(ki, vi, ko, vo, hist, nSel, shift);
        unsigned long long* tk = ki; ki = ko; ko = tk;
        unsigned* tv = vi; vi = vo; vo = tv;
    }
    // 6 passes: data ends back in keysA/valsA (== ki/vi after swaps)

    // 3) widen payload (original pos values) to int64 output
    cast_out_kernel<<<(nSel + 255) / 256, 256, 0, stream>>>(vi, (long long*)d_out, nSel);
}